// ModelNew_50388556316933
// MI455X (gfx1250) — compile-verified
//
#include <hip/hip_runtime.h>
#include <hip/hip_bf16.h>
#include <stdint.h>

typedef __attribute__((ext_vector_type(2))) float v2f;
typedef __attribute__((ext_vector_type(8))) float v8f;

#define B_STRIDE   786432    // 3 * 64^3
#define D_STRIDE   4096      // 64^2
#define NSPATIAL   238328.0f // 62^3

// ---------------- zero the 512-float workspace (256 sums + 256 sumsqs) -----
__global__ void zero_ws_kernel(float* ws) {
    int i = threadIdx.x;
    if (i < 512) ws[i] = 0.0f;
}

// ---------------- fused conv3d (+bias) + hardswish + partial sums ----------
// K reordered as k = kd*28 + inner, inner = ci*9+kh*3+kw (inner 27 = bias col
// in the kd=0 block, zero rows elsewhere). Per-lane K set is {khalf,khalf+1}
// mod 4 and 28 = 0 mod 4, so one depth-slice's per-lane values are reusable
// verbatim as the wave walks D: rolling 3-slice window, 14 loads per output.
__global__ __launch_bounds__(256) void conv_gn_fused_kernel(
    const float* __restrict__ x, const float* __restrict__ wgt,
    const float* __restrict__ cbias, float* __restrict__ ws)
{
    const int lane   = threadIdx.x & 31;
    const int bat    = blockIdx.y;
    const int waveId = blockIdx.x * 8 + (threadIdx.x >> 5);  // 0..495
    const int m      = lane & 15;                 // A-row (out channel)
    const int n      = lane & 15;                 // B-col (W position)
    const int khalf  = (lane >= 16) ? 2 : 0;      // K sub-pair per ISA layout
    const int mbase  = (lane >= 16) ? 8 : 0;      // C/D channel base

    // ---- A fragments in the permuted K-order -----------------------------
    float a0[21], a1[21];
#pragma unroll
    for (int s = 0; s < 21; ++s) {
        int k0 = 4 * s + khalf, k1 = k0 + 1;
        int kd0 = k0 / 28, in0 = k0 - kd0 * 28;
        int kd1 = k1 / 28, in1 = k1 - kd1 * 28;
        a0[s] = (in0 < 27)
                ? wgt[m * 81 + (in0 / 9) * 27 + kd0 * 9 + ((in0 % 9) / 3) * 3 + (in0 % 3)]
                : ((kd0 == 0) ? cbias[m] : 0.0f);
        a1[s] = (in1 < 27)
                ? wgt[m * 81 + (in1 / 9) * 27 + kd1 * 9 + ((in1 % 9) / 3) * 3 + (in1 % 3)]
                : ((kd1 == 0) ? cbias[m] : 0.0f);
    }

    // ---- per-lane slice offsets: offb[2p+e] for inner = 4p+khalf+e --------
    int offb[14];
#pragma unroll
    for (int p = 0; p < 7; ++p) {
#pragma unroll
        for (int e = 0; e < 2; ++e) {
            int in = 4 * p + khalf + e;
            if (in > 26) in = 26;                 // safe addr; value replaced
            int ci = in / 9, kh = (in % 9) / 3, kw = in % 3;
            offb[2 * p + e] = ci * 262144 + kh * 64 + kw;
        }
    }

    float sum8[8], sq8[8];
#pragma unroll
    for (int v = 0; v < 8; ++v) { sum8[v] = 0.0f; sq8[v] = 0.0f; }

    // ---- task decode: (h, wtile, d-segment) -------------------------------
    const int dseg = waveId & 1;
    const int rest = waveId >> 1;            // 0..247
    const int h    = rest >> 2;              // 0..61
    const int wt   = rest & 3;
    // tile 3 overlaps tile 2 (w0=46): max read = 46+15+2 = 63, no clamp;
    // the two duplicated positions (n<2) are masked out of the sums.
    const int w0   = (wt == 3) ? 46 : (wt << 4);
    const int d0   = dseg * 31;              // segments [0,31) and [31,62)
    const float pm = ((wt < 3) || (n >= 2)) ? 1.0f : 0.0f;

    const unsigned colbase = (unsigned)(bat * B_STRIDE + h * 64 + w0 + n);

    auto load_slice = [&](int dslice, float (&buf)[14]) {
        unsigned sb = colbase + (unsigned)dslice * D_STRIDE;
#pragma unroll
        for (int j = 0; j < 14; ++j) buf[j] = x[sb + (unsigned)offb[j]];
        // inner==27 slot (khalf=2, p=6, e=1): bias column B value = 1.0
        buf[13] = (khalf == 2) ? 1.0f : buf[13];   // select, no EXEC change
    };

    auto step = [&](int dd, float (&bA)[14], float (&bB)[14], float (&bC)[14]) {
        load_slice(dd + 2, bC);                // prefetch role-kd2 slice
        v8f c = {};
#pragma unroll
        for (int p = 0; p < 7; ++p) {          // kd=0 block (resident)
            v2f af, bf;
            af[0] = a0[p];      af[1] = a1[p];
            bf[0] = bA[2 * p];  bf[1] = bA[2 * p + 1];
            c = __builtin_amdgcn_wmma_f32_16x16x4_f32(
                    false, af, false, bf, (short)0, c, false, false);
        }
#pragma unroll
        for (int p = 0; p < 7; ++p) {          // kd=1 block (resident)
            v2f af, bf;
            af[0] = a0[7 + p];  af[1] = a1[7 + p];
            bf[0] = bB[2 * p];  bf[1] = bB[2 * p + 1];
            c = __builtin_amdgcn_wmma_f32_16x16x4_f32(
                    false, af, false, bf, (short)0, c, false, false);
        }
#pragma unroll
        for (int p = 0; p < 7; ++p) {          // kd=2 block (just loaded)
            v2f af, bf;
            af[0] = a0[14 + p]; af[1] = a1[14 + p];
            bf[0] = bC[2 * p];  bf[1] = bC[2 * p + 1];
            c = __builtin_amdgcn_wmma_f32_16x16x4_f32(
                    false, af, false, bf, (short)0, c, false, false);
        }
#pragma unroll
        for (int v = 0; v < 8; ++v) {
            float z  = c[v];                   // bias already folded in
            float cl = fminf(fmaxf(z + 3.0f, 0.0f), 6.0f);
            float hs = z * cl * (1.0f / 6.0f) * pm;
            sum8[v] += hs;
            sq8[v]  += hs * hs;
        }
    };

    // ---- rolling-window walk over 31 consecutive D outputs ----------------
    float bufA[14], bufB[14], bufC[14];
    load_slice(d0,     bufA);
    load_slice(d0 + 1, bufB);
    int d = d0;
#pragma unroll 1
    for (int it = 0; it < 10; ++it) {          // 30 outputs, buffers rotate
        step(d + 0, bufA, bufB, bufC);
        step(d + 1, bufB, bufC, bufA);
        step(d + 2, bufC, bufA, bufB);
        d += 3;
    }
    step(d, bufA, bufB, bufC);                 // 31st output

    // ---- reduce: butterfly within each 16-lane half (position dim) --------
    __shared__ float s_acc[32];                // [0..15] sums, [16..31] sumsqs
    if (threadIdx.x < 32) s_acc[threadIdx.x] = 0.0f;
    __syncthreads();

#pragma unroll
    for (int v = 0; v < 8; ++v) {
#pragma unroll
        for (int xm = 1; xm <= 8; xm <<= 1) {
            sum8[v] += __shfl_xor(sum8[v], xm, 32);
            sq8[v]  += __shfl_xor(sq8[v],  xm, 32);
        }
    }
    if ((lane & 15) == 0) {                    // lanes 0 (ch 0-7), 16 (ch 8-15)
#pragma unroll
        for (int v = 0; v < 8; ++v) {
            atomicAdd(&s_acc[mbase + v],      sum8[v]);   // ds_add_f32
            atomicAdd(&s_acc[16 + mbase + v], sq8[v]);
        }
    }
    __syncthreads();
    if (threadIdx.x < 32) {
        int cc    = threadIdx.x & 15;
        int which = threadIdx.x >> 4;          // 0 = sum, 1 = sumsq
        atomicAdd(&ws[which * 256 + bat * 16 + cc], s_acc[threadIdx.x]);
    }
}

// ---------------- finalize: GroupNorm stats + per-(b,c) spatial mean -------
__global__ void finalize_kernel(const float* __restrict__ ws,
                                const float* __restrict__ gw,
                                const float* __restrict__ gb,
                                float* __restrict__ out)
{
    int i = threadIdx.x;                 // 0..255
    int b = i >> 4, c = i & 15;
    int g = c >> 2;                      // 4 groups of 4 channels
    const float* S1 = ws;
    const float* S2 = ws + 256;
    float s1g = 0.0f, s2g = 0.0f;
#pragma unroll
    for (int j = 0; j < 4; ++j) {
        s1g += S1[b * 16 + g * 4 + j];
        s2g += S2[b * 16 + g * 4 + j];
    }
    float mu  = s1g / (4.0f * NSPATIAL);
    float e2  = s2g / (4.0f * NSPATIAL);
    float var = e2 - mu * mu;
    float inv = rsqrtf(var + 1e-5f);
    out[i] = (S1[b * 16 + c] / NSPATIAL - mu) * inv * gw[c] + gb[c];
}

extern "C" void kernel_launch(void* const* d_in, const int* in_sizes, int n_in,
                              void* d_out, int out_size, void* d_ws, size_t ws_size,
                              hipStream_t stream) {
    const float* x    = (const float*)d_in[0];  // (16,3,64,64,64)
    const float* wgt  = (const float*)d_in[1];  // (16,3,3,3,3)
    const float* cb   = (const float*)d_in[2];  // (16,)
    const float* gwp  = (const float*)d_in[3];  // (16,)
    const float* gbp  = (const float*)d_in[4];  // (16,)
    float* out = (float*)d_out;                 // (16,16) fp32
    float* ws  = (float*)d_ws;                  // 512 floats used

    zero_ws_kernel<<<dim3(1), dim3(512), 0, stream>>>(ws);
    conv_gn_fused_kernel<<<dim3(62, 16), dim3(256), 0, stream>>>(x, wgt, cb, ws);
    finalize_kernel<<<dim3(1), dim3(256), 0, stream>>>(ws, gwp, gbp, out);
}